// GNN_85366769975686
// MI455X (gfx1250) — compile-verified
//
#include <hip/hip_runtime.h>

typedef float v2f __attribute__((ext_vector_type(2)));
typedef float v8f __attribute__((ext_vector_type(8)));

// ---------------------------------------------------------------------------
// Zero the aggregation halves (cols 0..127) of both node tables.
// Each thread clears one float4 in each table.
// ---------------------------------------------------------------------------
__global__ void gnn_zero_agg(float* __restrict__ ws, float* __restrict__ out, int n_nodes) {
    long idx   = (long)blockIdx.x * blockDim.x + threadIdx.x;
    long total = (long)n_nodes * 32;            // 128 floats = 32 float4 per row
    if (idx >= total) return;
    long row = idx >> 5;
    long q   = idx & 31;
    float4 z = make_float4(0.f, 0.f, 0.f, 0.f);
    ((float4*)(ws  + row * 256))[q] = z;
    ((float4*)(out + row * 256))[q] = z;
}

// ---------------------------------------------------------------------------
// f32 WMMA GEMM:  C[:, col_off .. col_off+127] = A[M x K] * B[K x 128] + bias
//   - A row-major, lda == K
//   - C row-major with stride ldc (=256), output written at column offset
//   - one block = 256 threads = 8 wave32; block computes a 16 x 128 stripe
//   - per-wave 16x16 tile accumulated with V_WMMA_F32_16X16X4_F32 (exact f32)
// ---------------------------------------------------------------------------
#define KMAX 256
#define APAD 260   // 260 % 64 == 4 -> 16 rows hit 16 distinct LDS banks

__global__ void gnn_gemm_wmma(const float* __restrict__ A, int K,
                              const float* __restrict__ B,
                              const float* __restrict__ bias,
                              float* __restrict__ C, int ldc, int col_off) {
    __shared__ float As[16][APAD];

    const int tid   = threadIdx.x;
    const int mbase = blockIdx.x * 16;

    // Cooperative, coalesced load of the 16 x K A-stripe into LDS.
    for (int i = tid; i < 16 * K; i += 256) {
        int r = i / K, c = i % K;
        As[r][c] = A[(long)(mbase + r) * K + c];
    }
    __syncthreads();

    const int wave  = tid >> 5;          // 0..7  -> N-tile
    const int lane  = tid & 31;
    const int nbase = wave * 16;
    const int n     = lane & 15;         // column within tile
    const int hi    = lane >> 4;         // lane half selects K/M offset

    // Accumulator initialized with the bias (bias depends only on column).
    v8f acc;
    {
        float bv = bias[nbase + n];
        for (int g = 0; g < 8; ++g) acc[g] = bv;
    }

    // K loop: V_WMMA_F32_16X16X4_F32 consumes K=4 per issue.
    for (int k = 0; k < K; k += 4) {
        int kb = k + 2 * hi;
        v2f a, b;
        a.x = As[n][kb];                 // A[m = lane&15][kb]
        a.y = As[n][kb + 1];
        b.x = B[(long)kb       * 128 + nbase + n];   // B[kb][n]
        b.y = B[(long)(kb + 1) * 128 + nbase + n];
        acc = __builtin_amdgcn_wmma_f32_16x16x4_f32(
                  /*neg_a=*/false, a, /*neg_b=*/false, b,
                  /*c_mod=*/(short)0, acc, /*reuse_a=*/false, /*reuse_b=*/false);
    }

    // Store: VGPR g holds rows g (lanes 0-15) and g+8 (lanes 16-31).
    for (int g = 0; g < 8; ++g) {
        int row = mbase + g + 8 * hi;
        C[(long)row * ldc + col_off + nbase + n] = acc[g];
    }
}

// ---------------------------------------------------------------------------
// Edge scatter:  agg[dst[e]][0:128] += msg[src[e]][0:128]
// msg rows live at table + row*256 + 128, agg rows at table + row*256.
// One wave per edge; each lane handles 4 contiguous floats (float4 load,
// 4 x global_atomic_add_f32). Node tables fit in L2, so atomics stay on-chip.
// ---------------------------------------------------------------------------
__global__ void gnn_scatter(const int* __restrict__ src, const int* __restrict__ dst,
                            const float* __restrict__ msg_tab, float* __restrict__ agg_tab,
                            int n_edges) {
    int edge = blockIdx.x * 8 + (threadIdx.x >> 5);   // 8 waves per 256-thread block
    int lane = threadIdx.x & 31;
    if (edge >= n_edges) return;

    int s = src[edge];
    int d = dst[edge];

    const float4 v = ((const float4*)(msg_tab + (long)s * 256 + 128))[lane];
    float* a = agg_tab + (long)d * 256 + lane * 4;
    atomicAdd(a + 0, v.x);
    atomicAdd(a + 1, v.y);
    atomicAdd(a + 2, v.z);
    atomicAdd(a + 3, v.w);
}

// ---------------------------------------------------------------------------
// Launch: zero -> GEMM1 -> scatter1 -> GEMM2 -> scatter2
// ---------------------------------------------------------------------------
extern "C" void kernel_launch(void* const* d_in, const int* in_sizes, int n_in,
                              void* d_out, int out_size, void* d_ws, size_t ws_size,
                              hipStream_t stream) {
    const float* features = (const float*)d_in[0];   // [N,128]
    const int*   src      = (const int*)  d_in[1];   // [E]
    const int*   dst      = (const int*)  d_in[2];   // [E]
    const float* W1       = (const float*)d_in[3];   // [128,128]
    const float* b1       = (const float*)d_in[4];   // [128]
    const float* W2       = (const float*)d_in[5];   // [256,128]
    const float* b2       = (const float*)d_in[6];   // [128]

    float* out   = (float*)d_out;                    // [N,256]
    float* feat1 = (float*)d_ws;                     // [N,256] scratch (51.2 MB)

    const int N = in_sizes[0] / 128;                 // 50000 (multiple of 16)
    const int E = in_sizes[1];                       // 800000

    // 1) zero agg halves of both tables
    {
        long total  = (long)N * 32;
        int  blocks = (int)((total + 255) / 256);
        gnn_zero_agg<<<blocks, 256, 0, stream>>>(feat1, out, N);
    }

    // 2) m1 = features @ W1 + b1  -> feat1 cols 128..255
    gnn_gemm_wmma<<<N / 16, 256, 0, stream>>>(features, 128, W1, b1, feat1, 256, 128);

    // 3) agg1[dst] += m1[src]     -> feat1 cols 0..127
    {
        int blocks = (E + 7) / 8;
        gnn_scatter<<<blocks, 256, 0, stream>>>(src, dst, feat1, feat1, E);
    }

    // 4) m2 = feat1 @ W2 + b2     -> out cols 128..255
    gnn_gemm_wmma<<<N / 16, 256, 0, stream>>>(feat1, 256, W2, b2, out, 256, 128);

    // 5) agg2[dst] += m2[src]     -> out cols 0..127
    {
        int blocks = (E + 7) / 8;
        gnn_scatter<<<blocks, 256, 0, stream>>>(src, dst, out, out, E);
    }
}